// AdditiveAttention_46050639348290
// MI455X (gfx1250) — compile-verified
//
#include <hip/hip_runtime.h>
#include <math.h>

typedef __attribute__((ext_vector_type(2))) float v2f;
typedef __attribute__((ext_vector_type(8))) float v8f;

constexpr int B_ = 16, Q_ = 64, K_ = 512, H_ = 256, DV_ = 512, DQ_ = 512;
#define MASK_VAL_F (-1e6f)

// fp32 WMMA GEMM: C[M,N] = A[M,Kd] @ Bm[Kd,N], batched via blockIdx.y with
// flat element strides sA/sB/sC. Each wave owns a 64x16 output tile:
// per K-step the B fragment is loaded once and reused by 4 independent
// V_WMMA_F32_16X16X4_F32 ops (4 accumulators), so the matrix pipe sees
// 4 WMMAs per (4 x b64 + 2 x b32) loads. M % 64 == 0, N % 16 == 0, Kd % 4 == 0.
__global__ __launch_bounds__(128) void wmma_gemm_f32_kernel(
    const float* __restrict__ A, const float* __restrict__ Bm,
    float* __restrict__ C, int M, int N, int Kd,
    long sA, long sB, long sC)
{
    const int batch = blockIdx.y;
    const float* Ab = A  + (size_t)batch * (size_t)sA;
    const float* Bb = Bm + (size_t)batch * (size_t)sB;
    float*       Cb = C  + (size_t)batch * (size_t)sC;

    const int wave = threadIdx.x >> 5;
    const int lane = threadIdx.x & 31;
    const int half = lane >> 4;   // 0: lanes 0-15 hold K=0,1; 1: lanes 16-31 hold K=2,3
    const int l16  = lane & 15;

    const int tile = blockIdx.x * 4 + wave;
    const int ntn  = N >> 4;           // 16-wide column tiles
    const int tm   = tile / ntn;       // 64-row block index
    const int tn   = tile % ntn;

    const size_t Kz = (size_t)Kd, Nz = (size_t)N;
    // A fragment rows: M = tm*64 + sub*16 + l16; lane holds K = k0+2*half, +1
    const float* Arow = Ab + (size_t)(tm * 64 + l16) * Kz;
    // B fragment col:  N = tn*16 + l16; same K pair per lane-half
    const float* Bcol = Bb + (size_t)(tn * 16 + l16);

    v8f acc0 = {0.f,0.f,0.f,0.f,0.f,0.f,0.f,0.f};
    v8f acc1 = acc0, acc2 = acc0, acc3 = acc0;

    for (int k0 = 0; k0 < Kd; k0 += 4) {
        const int ka = k0 + 2 * half;
        v2f b;
        b.x = Bcol[(size_t)ka * Nz];        // B[ka  ][n]
        b.y = Bcol[(size_t)(ka + 1) * Nz];  // B[ka+1][n]
        v2f a0 = *(const v2f*)(Arow + ka);
        v2f a1 = *(const v2f*)(Arow + 16 * Kz + ka);
        v2f a2 = *(const v2f*)(Arow + 32 * Kz + ka);
        v2f a3 = *(const v2f*)(Arow + 48 * Kz + ka);
        acc0 = __builtin_amdgcn_wmma_f32_16x16x4_f32(false, a0, false, b, (short)0, acc0, false, false);
        acc1 = __builtin_amdgcn_wmma_f32_16x16x4_f32(false, a1, false, b, (short)0, acc1, false, false);
        acc2 = __builtin_amdgcn_wmma_f32_16x16x4_f32(false, a2, false, b, (short)0, acc2, false, false);
        acc3 = __builtin_amdgcn_wmma_f32_16x16x4_f32(false, a3, false, b, (short)0, acc3, false, false);
    }

    // C/D layout: VGPR r = row (base + r + 8*half), col (tn*16 + l16)
    float* Crow = Cb + (size_t)(tm * 64 + 8 * half) * Nz + (tn * 16 + l16);
    #pragma unroll
    for (int r = 0; r < 8; ++r) Crow[(size_t)r * Nz] = acc0[r];
    Crow += 16 * Nz;
    #pragma unroll
    for (int r = 0; r < 8; ++r) Crow[(size_t)r * Nz] = acc1[r];
    Crow += 16 * Nz;
    #pragma unroll
    for (int r = 0; r < 8; ++r) Crow[(size_t)r * Nz] = acc2[r];
    Crow += 16 * Nz;
    #pragma unroll
    for (int r = 0; r < 8; ++r) Crow[(size_t)r * Nz] = acc3[r];
}

// One block per (b,q). 8 waves; wave w owns keys k = w, w+8, ...
// score[k] = sum_h Wv[h] * tanh(qh[b,q,h] + kh[b,k,h]); then masked softmax
// over the K=512 axis with 256 threads (2 entries per thread).
__global__ __launch_bounds__(256) void score_softmax_kernel(
    const float* __restrict__ qh, const float* __restrict__ kh,
    const float* __restrict__ Wv, const int* __restrict__ valid_lens,
    float* __restrict__ attn)
{
    __shared__ float s_q[H_];
    __shared__ float s_w[H_];
    __shared__ float s_sc[K_];
    __shared__ float s_red[256];

    const int b   = blockIdx.x / Q_;
    const int q   = blockIdx.x % Q_;
    const int tid = threadIdx.x;

    s_q[tid] = qh[((size_t)b * Q_ + q) * H_ + tid];
    s_w[tid] = Wv[tid];
    __syncthreads();

    const int wave = tid >> 5, lane = tid & 31;
    for (int k = wave; k < K_; k += 8) {
        const float* krow = kh + ((size_t)b * K_ + k) * H_;
        float s = 0.f;
        #pragma unroll
        for (int h = lane; h < H_; h += 32)
            s += s_w[h] * tanhf(s_q[h] + krow[h]);
        #pragma unroll
        for (int off = 16; off > 0; off >>= 1)      // wave32 reduction
            s += __shfl_down(s, off, 32);
        if (lane == 0) s_sc[k] = s;
    }
    __syncthreads();

    const int vlen = valid_lens[b];
    float v0 = (tid       < vlen) ? s_sc[tid]       : MASK_VAL_F;
    float v1 = (tid + 256 < vlen) ? s_sc[tid + 256] : MASK_VAL_F;

    // block-wide max
    s_red[tid] = fmaxf(v0, v1);
    __syncthreads();
    for (int off = 128; off > 0; off >>= 1) {
        if (tid < off) s_red[tid] = fmaxf(s_red[tid], s_red[tid + off]);
        __syncthreads();
    }
    const float mx = s_red[0];
    __syncthreads();

    const float e0 = expf(v0 - mx);
    const float e1 = expf(v1 - mx);
    s_red[tid] = e0 + e1;
    __syncthreads();
    for (int off = 128; off > 0; off >>= 1) {
        if (tid < off) s_red[tid] += s_red[tid + off];
        __syncthreads();
    }
    const float inv = 1.0f / s_red[0];

    const size_t base = ((size_t)b * Q_ + q) * K_;
    attn[base + tid]       = e0 * inv;
    attn[base + tid + 256] = e1 * inv;
}

extern "C" void kernel_launch(void* const* d_in, const int* in_sizes, int n_in,
                              void* d_out, int out_size, void* d_ws, size_t ws_size,
                              hipStream_t stream)
{
    const float* queries = (const float*)d_in[0];  // [B,Q,DQ]
    const float* keys    = (const float*)d_in[1];  // [B,K,DK]
    const float* values  = (const float*)d_in[2];  // [B,K,DV]
    const int*   vlens   = (const int*)  d_in[3];  // [B]
    const float* Wq      = (const float*)d_in[4];  // [DQ,H]
    const float* Wk      = (const float*)d_in[5];  // [DK,H]
    const float* Wv      = (const float*)d_in[6];  // [H]
    float* out = (float*)d_out;                    // [B,Q,DV]

    char* ws = (char*)d_ws;
    float* qh   = (float*)ws;                                          // B*Q*H
    float* kh   = (float*)(ws + (size_t)B_ * Q_ * H_ * sizeof(float)); // B*K*H
    float* attn = (float*)(ws + (size_t)B_ * Q_ * H_ * sizeof(float)
                              + (size_t)B_ * K_ * H_ * sizeof(float)); // B*Q*K

    // 1) q projection: [B*Q, DQ] @ [DQ, H]
    {
        const int M = B_ * Q_, N = H_, Kd = DQ_;
        const int tiles = (M / 64) * (N / 16);     // 16*16 = 256
        wmma_gemm_f32_kernel<<<dim3(tiles / 4, 1), 128, 0, stream>>>(
            queries, Wq, qh, M, N, Kd, 0, 0, 0);
    }
    // 2) k projection: [B*K, DK] @ [DK, H]
    {
        const int M = B_ * K_, N = H_, Kd = DQ_;
        const int tiles = (M / 64) * (N / 16);     // 128*16 = 2048
        wmma_gemm_f32_kernel<<<dim3(tiles / 4, 1), 128, 0, stream>>>(
            keys, Wk, kh, M, N, Kd, 0, 0, 0);
    }
    // 3) scores + masked softmax -> attn
    score_softmax_kernel<<<dim3(B_ * Q_), 256, 0, stream>>>(qh, kh, Wv, vlens, attn);
    // 4) out[b] = attn[b] @ values[b]   (batched over blockIdx.y)
    {
        const int M = Q_, N = DV_, Kd = K_;
        const int tiles = (M / 64) * (N / 16);     // 1*32 = 32
        wmma_gemm_f32_kernel<<<dim3(tiles / 4, B_), 128, 0, stream>>>(
            attn, values, out, M, N, Kd,
            (long)Q_ * K_, (long)K_ * DV_, (long)Q_ * DV_);
    }
}